// BLAMem_60430189855124
// MI455X (gfx1250) — compile-verified
//
#include <hip/hip_runtime.h>
#include <hip/hip_bf16.h>
#include <math.h>

// Problem constants (fixed by the reference setup_inputs)
#define NBATCH 32
#define TLEN   4096
#define NCHUNK 256
#define CHUNKL 16
#define C1 9
#define C2 81
#define C3 729
#define MDIM 819      // 9 + 81 + 729
#define HDIM 256
#define KPAD 820      // MDIM padded to multiple of 4 for wmma f32 16x16x4
#define LNEPS 1e-5f

typedef __attribute__((ext_vector_type(2))) float v2f;
typedef __attribute__((ext_vector_type(8))) float v8f;

__device__ __forceinline__ float wave_sum32(float v) {
#pragma unroll
  for (int o = 16; o > 0; o >>= 1) v += __shfl_xor(v, o, 32);
  return v;
}

__device__ __forceinline__ float gelu_exact(float x) {
  return 0.5f * x * (1.0f + erff(x * 0.7071067811865475f));
}

// ---------------------------------------------------------------------------
// Kernel 1: per-chunk depth-3 log-signature + per-level layernorm.
// One wave (32 threads) per (batch, chunk). Level-3 state (729) lives in
// 23 regs/lane; levels 1-2 live in LDS for cross-lane broadcast.
// sig <- sig * exp(dx) per step:   (a0 = b0 = 1 throughout)
//   new3[ijk] = a3 + a2[ij] v[k] + a1[i] v[j]v[k]/2 + v[i]v[j]v[k]/6
//   new2[jk]  = a2 + a1[j] v[k] + v[j]v[k]/2
//   new1[j]   = a1 + v[j]
// log (m0 = 0):
//   l1 = m1 ; l2 = m2 - m1⊗m1/2 ;
//   l3 = m3 - (m1⊗m2 + m2⊗m1)/2 + m1⊗m1⊗m1/3
// ---------------------------------------------------------------------------
__global__ __launch_bounds__(32)
void chunk_logsig_kernel(const float* __restrict__ x,
                         const float* __restrict__ g1, const float* __restrict__ b1,
                         const float* __restrict__ g2, const float* __restrict__ b2,
                         const float* __restrict__ g3, const float* __restrict__ b3,
                         float* __restrict__ out) {
  __shared__ float s1[C1];
  __shared__ float s2[C2];
  __shared__ float sv[C1];
  const int blk  = blockIdx.x;       // b * NCHUNK + n
  const int bb   = blk / NCHUNK;
  const int n    = blk % NCHUNK;
  const int lane = threadIdx.x;

  float a3[23];
#pragma unroll
  for (int r = 0; r < 23; ++r) a3[r] = 0.0f;
  if (lane < C1) s1[lane] = 0.0f;
#pragma unroll
  for (int q = 0; q < 3; ++q) { int e = lane + 32 * q; if (e < C2) s2[e] = 0.0f; }
  __syncthreads();

  for (int t = 0; t < CHUNKL; ++t) {
    const float* xp = x + ((size_t)bb * TLEN + (size_t)n * CHUNKL + (size_t)t) * 8;
    if (lane < 8) sv[lane] = (t == 0) ? xp[lane] : (xp[lane] - xp[lane - 8]);
    if (lane == 8) sv[8] = (t == 0) ? ((float)(n * CHUNKL) / (float)TLEN)
                                    : (1.0f / (float)TLEN);
    __syncthreads();

    float n3[23];
#pragma unroll
    for (int r = 0; r < 23; ++r) {
      int e = lane + 32 * r;
      if (e < C3) {
        int i = e / 81, j = (e / 9) % 9, k = e % 9;
        float vj = sv[j], vk = sv[k];
        n3[r] = a3[r] + s2[i * 9 + j] * vk + s1[i] * (0.5f * vj * vk)
              + (1.0f / 6.0f) * sv[i] * vj * vk;
      } else n3[r] = 0.0f;
    }
    float n2[3];
#pragma unroll
    for (int q = 0; q < 3; ++q) {
      int e = lane + 32 * q;
      if (e < C2) {
        int j = e / 9, k = e % 9;
        n2[q] = s2[e] + s1[j] * sv[k] + 0.5f * sv[j] * sv[k];
      } else n2[q] = 0.0f;
    }
    float n1 = (lane < C1) ? (s1[lane] + sv[lane]) : 0.0f;
    __syncthreads();
    if (lane < C1) s1[lane] = n1;
#pragma unroll
    for (int q = 0; q < 3; ++q) { int e = lane + 32 * q; if (e < C2) s2[e] = n2[q]; }
    __syncthreads();
#pragma unroll
    for (int r = 0; r < 23; ++r) a3[r] = n3[r];
  }

  // log-signature (reads s1,s2; no LDS writes, so no barrier needed)
  float l2[3];
#pragma unroll
  for (int q = 0; q < 3; ++q) {
    int e = lane + 32 * q;
    if (e < C2) { int j = e / 9, k = e % 9; l2[q] = s2[e] - 0.5f * s1[j] * s1[k]; }
    else l2[q] = 0.0f;
  }
  float l1 = (lane < C1) ? s1[lane] : 0.0f;
#pragma unroll
  for (int r = 0; r < 23; ++r) {
    int e = lane + 32 * r;
    if (e < C3) {
      int i = e / 81, j = (e / 9) % 9, k = e % 9;
      a3[r] = a3[r]
            - 0.5f * (s1[i] * s2[j * 9 + k] + s2[i * 9 + j] * s1[k])
            + (1.0f / 3.0f) * s1[i] * s1[j] * s1[k];
    }
  }

  float* op = out + (size_t)blk * MDIM;
  // layernorm level 1 (dim 9)
  {
    float mu = wave_sum32(l1) * (1.0f / C1);
    float d  = (lane < C1) ? (l1 - mu) : 0.0f;
    float rs = rsqrtf(wave_sum32(d * d) * (1.0f / C1) + LNEPS);
    if (lane < C1) op[lane] = d * rs * g1[lane] + b1[lane];
  }
  // layernorm level 2 (dim 81)
  {
    float s = 0.0f;
#pragma unroll
    for (int q = 0; q < 3; ++q) { int e = lane + 32 * q; if (e < C2) s += l2[q]; }
    float mu = wave_sum32(s) * (1.0f / C2);
    float vs = 0.0f;
#pragma unroll
    for (int q = 0; q < 3; ++q) {
      int e = lane + 32 * q;
      if (e < C2) { float d = l2[q] - mu; vs += d * d; }
    }
    float rs = rsqrtf(wave_sum32(vs) * (1.0f / C2) + LNEPS);
#pragma unroll
    for (int q = 0; q < 3; ++q) {
      int e = lane + 32 * q;
      if (e < C2) op[C1 + e] = (l2[q] - mu) * rs * g2[e] + b2[e];
    }
  }
  // layernorm level 3 (dim 729)
  {
    float s = 0.0f;
#pragma unroll
    for (int r = 0; r < 23; ++r) { int e = lane + 32 * r; if (e < C3) s += a3[r]; }
    float mu = wave_sum32(s) * (1.0f / C3);
    float vs = 0.0f;
#pragma unroll
    for (int r = 0; r < 23; ++r) {
      int e = lane + 32 * r;
      if (e < C3) { float d = a3[r] - mu; vs += d * d; }
    }
    float rs = rsqrtf(wave_sum32(vs) * (1.0f / C3) + LNEPS);
#pragma unroll
    for (int r = 0; r < 23; ++r) {
      int e = lane + 32 * r;
      if (e < C3) op[C1 + C2 + e] = (a3[r] - mu) * rs * g3[e] + b3[e];
    }
  }
}

// --------------------------- tensor-algebra helpers ------------------------
__device__ __forceinline__ float exp3_elem(const float* __restrict__ L3g,
                                           const float* U1, const float* U2, int e) {
  int i = e / 81, j = (e / 9) % 9, k = e % 9;
  return L3g[e] + 0.5f * (U1[i] * U2[e % 81] + U2[e / 9] * U1[k])
       + (1.0f / 6.0f) * U1[i] * U1[j] * U1[k];
}
__device__ __forceinline__ float mul3_elem(const float* A1, const float* A2,
                                           const float* A3,
                                           const float* B1, const float* B2,
                                           float b3, int e) {
  int i = e / 81, k = e % 9;
  return A3[e] + A2[e / 9] * B1[k] + A1[i] * B2[e % 81] + b3;
}
__device__ __forceinline__ float log3_elem(const float* A1, const float* A2,
                                           const float* A3, int e) {
  int i = e / 81, j = (e / 9) % 9, k = e % 9;
  return A3[e] - 0.5f * (A1[i] * A2[e % 81] + A2[e / 9] * A1[k])
       + (1.0f / 3.0f) * A1[i] * A1[j] * A1[k];
}

// ---------------------------------------------------------------------------
// Kernel 2: reduce 8 consecutive chunk log-sigs into one group element.
// acc <- acc * exp(l_c), acc starts at identity (levels 1-3 zero, level0=1).
// ---------------------------------------------------------------------------
__global__ __launch_bounds__(256)
void reduce8_kernel(const float* __restrict__ logs, float* __restrict__ part) {
  __shared__ float A1[C1], A2[C2], A3[C3];
  __shared__ float Bb1[C1], Bb2[C2], Bb3[C3];
  __shared__ float U1[C1], U2[C2];
  const int tid = threadIdx.x;
  const int bb  = blockIdx.x / 32;
  const int g   = blockIdx.x % 32;

  if (tid < C1) A1[tid] = 0.0f;
  if (tid < C2) A2[tid] = 0.0f;
  A3[tid] = 0.0f; A3[tid + 256] = 0.0f;
  if (tid + 512 < C3) A3[tid + 512] = 0.0f;
  __syncthreads();

  for (int c = 0; c < 8; ++c) {
    const float* L = logs + (size_t)(bb * NCHUNK + g * 8 + c) * MDIM;
    if (tid < C1) U1[tid] = L[tid];
    if (tid < C2) U2[tid] = L[C1 + tid];
    __syncthreads();
    if (tid < C1) Bb1[tid] = U1[tid];
    if (tid < C2) { int j = tid / 9, k = tid % 9; Bb2[tid] = U2[tid] + 0.5f * U1[j] * U1[k]; }
    Bb3[tid]       = exp3_elem(L + C1 + C2, U1, U2, tid);
    Bb3[tid + 256] = exp3_elem(L + C1 + C2, U1, U2, tid + 256);
    if (tid + 512 < C3) Bb3[tid + 512] = exp3_elem(L + C1 + C2, U1, U2, tid + 512);
    __syncthreads();

    float m3a = mul3_elem(A1, A2, A3, Bb1, Bb2, Bb3[tid], tid);
    float m3b = mul3_elem(A1, A2, A3, Bb1, Bb2, Bb3[tid + 256], tid + 256);
    float m3c = 0.0f;
    if (tid + 512 < C3) m3c = mul3_elem(A1, A2, A3, Bb1, Bb2, Bb3[tid + 512], tid + 512);
    float m2 = 0.0f, m1 = 0.0f;
    if (tid < C2) { int j = tid / 9, k = tid % 9; m2 = A2[tid] + A1[j] * Bb1[k] + Bb2[tid]; }
    if (tid < C1) m1 = A1[tid] + Bb1[tid];
    __syncthreads();
    A3[tid] = m3a; A3[tid + 256] = m3b;
    if (tid + 512 < C3) A3[tid + 512] = m3c;
    if (tid < C2) A2[tid] = m2;
    if (tid < C1) A1[tid] = m1;
    __syncthreads();
  }

  float* P = part + (size_t)blockIdx.x * MDIM;
  if (tid < C1) P[tid] = A1[tid];
  if (tid < C2) P[C1 + tid] = A2[tid];
  P[C1 + C2 + tid] = A3[tid];
  P[C1 + C2 + tid + 256] = A3[tid + 256];
  if (tid + 512 < C3) P[C1 + C2 + tid + 512] = A3[tid + 512];
}

// ---------------------------------------------------------------------------
// Kernel 3: per batch, multiply 32 partial group elements in order, then log.
// ---------------------------------------------------------------------------
__global__ __launch_bounds__(256)
void reduce32_log_kernel(const float* __restrict__ part, float* __restrict__ pooled) {
  __shared__ float A1[C1], A2[C2], A3[C3];
  __shared__ float Bb1[C1], Bb2[C2];
  const int tid = threadIdx.x;
  const int bb  = blockIdx.x;

  if (tid < C1) A1[tid] = 0.0f;
  if (tid < C2) A2[tid] = 0.0f;
  A3[tid] = 0.0f; A3[tid + 256] = 0.0f;
  if (tid + 512 < C3) A3[tid + 512] = 0.0f;
  __syncthreads();

  for (int p = 0; p < 32; ++p) {
    const float* P = part + (size_t)(bb * 32 + p) * MDIM;
    if (tid < C1) Bb1[tid] = P[tid];
    if (tid < C2) Bb2[tid] = P[C1 + tid];
    __syncthreads();
    const float* P3 = P + C1 + C2;
    float m3a = mul3_elem(A1, A2, A3, Bb1, Bb2, P3[tid], tid);
    float m3b = mul3_elem(A1, A2, A3, Bb1, Bb2, P3[tid + 256], tid + 256);
    float m3c = 0.0f;
    if (tid + 512 < C3) m3c = mul3_elem(A1, A2, A3, Bb1, Bb2, P3[tid + 512], tid + 512);
    float m2 = 0.0f, m1 = 0.0f;
    if (tid < C2) { int j = tid / 9, k = tid % 9; m2 = A2[tid] + A1[j] * Bb1[k] + Bb2[tid]; }
    if (tid < C1) m1 = A1[tid] + Bb1[tid];
    __syncthreads();
    A3[tid] = m3a; A3[tid + 256] = m3b;
    if (tid + 512 < C3) A3[tid + 512] = m3c;
    if (tid < C2) A2[tid] = m2;
    if (tid < C1) A1[tid] = m1;
    __syncthreads();
  }

  float* O = pooled + (size_t)bb * MDIM;
  if (tid < C1) O[tid] = A1[tid];
  if (tid < C2) { int j = tid / 9, k = tid % 9; O[C1 + tid] = A2[tid] - 0.5f * A1[j] * A1[k]; }
  O[C1 + C2 + tid]       = log3_elem(A1, A2, A3, tid);
  O[C1 + C2 + tid + 256] = log3_elem(A1, A2, A3, tid + 256);
  if (tid + 512 < C3)
    O[C1 + C2 + tid + 512] = log3_elem(A1, A2, A3, tid + 512);
}

// ---------------------------------------------------------------------------
// Row layernorm: one block per row; output row stride S >= D, zero-padded.
// ---------------------------------------------------------------------------
__global__ __launch_bounds__(256)
void ln_row_kernel(const float* __restrict__ X, int D, int S,
                   const float* __restrict__ g, const float* __restrict__ bb,
                   float* __restrict__ Y) {
  __shared__ float red[256];
  const int tid = threadIdx.x;
  const int row = blockIdx.x;
  const float* xr = X + (size_t)row * D;
  float s = 0.0f;
  for (int e = tid; e < D; e += 256) s += xr[e];
  red[tid] = s; __syncthreads();
  for (int st = 128; st > 0; st >>= 1) { if (tid < st) red[tid] += red[tid + st]; __syncthreads(); }
  float mu = red[0] / (float)D; __syncthreads();
  float vs = 0.0f;
  for (int e = tid; e < D; e += 256) { float d = xr[e] - mu; vs += d * d; }
  red[tid] = vs; __syncthreads();
  for (int st = 128; st > 0; st >>= 1) { if (tid < st) red[tid] += red[tid + st]; __syncthreads(); }
  float rs = rsqrtf(red[0] / (float)D + LNEPS); __syncthreads();
  float* yr = Y + (size_t)row * S;
  for (int e = tid; e < S; e += 256)
    yr[e] = (e < D) ? ((xr[e] - mu) * rs * g[e] + bb[e]) : 0.0f;
}

// Pad W_in (819x256) to (820x256) with a zero last row.
__global__ void pad_win_kernel(const float* __restrict__ W, float* __restrict__ Wp) {
  int idx = blockIdx.x * blockDim.x + threadIdx.x;
  if (idx < KPAD * HDIM) {
    int row = idx / HDIM;
    Wp[idx] = (row < MDIM) ? W[idx] : 0.0f;
  }
}

// ---------------------------------------------------------------------------
// WMMA f32 GEMM: C = gelu?(A@B + bias) + resid. One wave per 16x16 tile.
// A row-major MxK (lda=K), B row-major KxN (ldb=N). K multiple of 4.
// Fragment layout per CDNA5 ISA 7.12.2 (16x16x4 f32):
//   A: lane<16 -> M=lane, {K=k0,k0+1}; lane>=16 -> M=lane-16, {K=k0+2,k0+3}
//   B: mirrored over K; C/D: vgpr r -> M = r + 8*(lane>=16), N = lane&15.
// ---------------------------------------------------------------------------
__global__ __launch_bounds__(32)
void wmma_gemm_kernel(const float* __restrict__ A, int lda,
                      const float* __restrict__ Bm, int ldb,
                      const float* __restrict__ bias,
                      const float* __restrict__ resid,
                      float* __restrict__ C, int ldc,
                      int K, int apply_gelu) {
  const int n0 = blockIdx.x * 16;
  const int m0 = blockIdx.y * 16;
  const int lane = threadIdx.x & 31;
  const int half = lane >> 4;
  const int lr   = lane & 15;

  v8f acc = {0.f, 0.f, 0.f, 0.f, 0.f, 0.f, 0.f, 0.f};
  const float* ap = A  + (size_t)(m0 + lr) * lda + 2 * half;
  const float* bp = Bm + (size_t)(2 * half) * ldb + n0 + lr;
  for (int k0 = 0; k0 < K; k0 += 4) {
    v2f av, bv;
    av.x = ap[k0];
    av.y = ap[k0 + 1];
    bv.x = bp[(size_t)k0 * ldb];
    bv.y = bp[(size_t)(k0 + 1) * ldb];
    acc = __builtin_amdgcn_wmma_f32_16x16x4_f32(false, av, false, bv,
                                                (short)0, acc, false, false);
  }
#pragma unroll
  for (int r = 0; r < 8; ++r) {
    int row = m0 + r + 8 * half;
    int col = n0 + lr;
    float v = acc[r] + bias[col];
    if (apply_gelu) v = gelu_exact(v);
    if (resid) v += resid[(size_t)row * ldc + col];
    C[(size_t)row * ldc + col] = v;
  }
}

// Final projection: out[b] = dot(h2[b,:], W_out[:,0]) + b_out
__global__ __launch_bounds__(256)
void out_kernel(const float* __restrict__ h2, const float* __restrict__ Wout,
                const float* __restrict__ bout, float* __restrict__ out) {
  __shared__ float red[256];
  const int tid = threadIdx.x;
  const int bb  = blockIdx.x;
  float s = h2[(size_t)bb * HDIM + tid] * Wout[tid];
  red[tid] = s; __syncthreads();
  for (int st = 128; st > 0; st >>= 1) { if (tid < st) red[tid] += red[tid + st]; __syncthreads(); }
  if (tid == 0) out[bb] = red[0] + bout[0];
}

// ---------------------------------------------------------------------------
extern "C" void kernel_launch(void* const* d_in, const int* in_sizes, int n_in,
                              void* d_out, int out_size, void* d_ws, size_t ws_size,
                              hipStream_t stream) {
  (void)in_sizes; (void)n_in; (void)out_size; (void)ws_size;
  const float* x        = (const float*)d_in[0];
  const float* ln_g1    = (const float*)d_in[1];
  const float* ln_b1    = (const float*)d_in[2];
  const float* ln_g2    = (const float*)d_in[3];
  const float* ln_b2    = (const float*)d_in[4];
  const float* ln_g3    = (const float*)d_in[5];
  const float* ln_b3    = (const float*)d_in[6];
  const float* in_ln_g  = (const float*)d_in[7];
  const float* in_ln_b  = (const float*)d_in[8];
  const float* W_in     = (const float*)d_in[9];
  const float* b_in     = (const float*)d_in[10];
  const float* res_ln_g = (const float*)d_in[11];
  const float* res_ln_b = (const float*)d_in[12];
  const float* W_r1     = (const float*)d_in[13];
  const float* b_r1     = (const float*)d_in[14];
  const float* W_r2     = (const float*)d_in[15];
  const float* b_r2     = (const float*)d_in[16];
  const float* W_out    = (const float*)d_in[17];
  const float* b_out    = (const float*)d_in[18];
  float* out = (float*)d_out;
  float* ws  = (float*)d_ws;

  float* ws_norm = ws;                                        // 8192 * 819
  float* ws_part = ws_norm + (size_t)NBATCH * NCHUNK * MDIM;  // 1024 * 819
  float* ws_pool = ws_part + (size_t)NBATCH * 32 * MDIM;      //   32 * 819
  float* A_pad   = ws_pool + (size_t)NBATCH * MDIM;           //   32 * 820
  float* B_pad   = A_pad   + (size_t)NBATCH * KPAD;           //  820 * 256
  float* h       = B_pad   + (size_t)KPAD * HDIM;             //   32 * 256
  float* r       = h  + (size_t)NBATCH * HDIM;
  float* r2      = r  + (size_t)NBATCH * HDIM;
  float* h2      = r2 + (size_t)NBATCH * HDIM;

  // 1) chunk log-signatures + per-level layernorm
  chunk_logsig_kernel<<<NBATCH * NCHUNK, 32, 0, stream>>>(
      x, ln_g1, ln_b1, ln_g2, ln_b2, ln_g3, ln_b3, ws_norm);

  // 2) BCH reduction (only last prefix element is needed): exp + t_mul tree
  reduce8_kernel<<<NBATCH * 32, 256, 0, stream>>>(ws_norm, ws_part);
  reduce32_log_kernel<<<NBATCH, 256, 0, stream>>>(ws_part, ws_pool);

  // 3) MLP head (WMMA f32 path)
  ln_row_kernel<<<NBATCH, 256, 0, stream>>>(ws_pool, MDIM, KPAD, in_ln_g, in_ln_b, A_pad);
  pad_win_kernel<<<(KPAD * HDIM + 255) / 256, 256, 0, stream>>>(W_in, B_pad);

  dim3 gtiles(HDIM / 16, NBATCH / 16);
  wmma_gemm_kernel<<<gtiles, 32, 0, stream>>>(A_pad, KPAD, B_pad, HDIM, b_in,
                                              nullptr, h, HDIM, KPAD, 1);
  ln_row_kernel<<<NBATCH, 256, 0, stream>>>(h, HDIM, HDIM, res_ln_g, res_ln_b, r);
  wmma_gemm_kernel<<<gtiles, 32, 0, stream>>>(r, HDIM, W_r1, HDIM, b_r1,
                                              nullptr, r2, HDIM, HDIM, 1);
  wmma_gemm_kernel<<<gtiles, 32, 0, stream>>>(r2, HDIM, W_r2, HDIM, b_r2,
                                              h, h2, HDIM, HDIM, 0);
  out_kernel<<<NBATCH, 256, 0, stream>>>(h2, W_out, b_out, out);
}